// CornerDecoder_22780506538601
// MI455X (gfx1250) — compile-verified
//
#include <hip/hip_runtime.h>
#include <math.h>

typedef __attribute__((ext_vector_type(2))) float v2f;
typedef __attribute__((ext_vector_type(8))) float v8f;

#define KTOP 100
#define NDETS 1000
#define CC 80
#define HH 384
#define WW 384
#define HW (HH*WW)
#define ROWS 32
#define SEGW 48
#define CAND_CAP 6144
#define NBINS 2048
#define BIN_SHIFT 21
#define SORT_N 16384
#define NCHUNK ((ROWS + 2) * (WW / 4))   /* 16B chunks per tile */

// monotone float <-> sortable-u32 mapping
__device__ __forceinline__ unsigned fkey(float f) {
    unsigned b = __float_as_uint(f);
    return (b & 0x80000000u) ? ~b : (b | 0x80000000u);
}
__device__ __forceinline__ float kfloat(unsigned k) {
    unsigned b = (k & 0x80000000u) ? (k ^ 0x80000000u) : ~k;
    return __uint_as_float(b);
}

struct Cand { unsigned key, idx; };

// Async tile loader: 34 rows x 384 cols of one channel into LDS using
// CDNA5 global_load_async_to_lds_b128 (GVS: SGPR base + per-lane u32 offset,
// per-lane LDS byte address). Out-of-image halo chunks get -inf via plain
// LDS stores to disjoint addresses. Caller must __syncthreads() after.
__device__ __forceinline__ void load_tile_async(float* tile, const float* cbase,
                                                int r0, int tid) {
    unsigned lbase = (unsigned)(size_t)tile;   // low 32 bits = LDS byte offset
    for (int i = tid; i < NCHUNK; i += 256) {
        int tr = i / (WW / 4);
        int c16 = i - tr * (WW / 4);
        int gr = r0 - 1 + tr;
        if (gr >= 0 && gr < HH) {
            unsigned goff = (unsigned)gr * (WW * 4) + (unsigned)c16 * 16u;
            unsigned laddr = lbase + (unsigned)i * 16u;
            asm volatile("global_load_async_to_lds_b128 %0, %1, %2"
                         :: "v"(laddr), "v"(goff), "s"(cbase) : "memory");
        } else {
            float4 v;
            v.x = v.y = v.z = v.w = -__builtin_inff();
            ((float4*)tile)[i] = v;
        }
    }
    asm volatile("s_wait_asynccnt 0x0" ::: "memory");
}

// ---------------------------------------------------------------------------
// Pass 1: fused sigmoid-NMS (monotone -> decide on raw logits) + 2048-bin
// histogram of sortable keys of surviving peaks. Sliding 3x3 window:
// keep(x) <=> x >= max9 of window, max9 built from rolling column max3.
// ---------------------------------------------------------------------------
__global__ __launch_bounds__(256) void nms_hist(const float* __restrict__ tlh,
                                                const float* __restrict__ brh,
                                                unsigned* __restrict__ hist) {
    __shared__ __align__(16) float tile[(ROWS + 2) * WW];
    __shared__ unsigned shist[NBINS];
    const int tid = threadIdx.x;
    const int z = blockIdx.z;
    const float* heat = z ? brh : tlh;
    const int c = blockIdx.y;
    const int r0 = blockIdx.x * ROWS;

    for (int i = tid; i < NBINS; i += 256) shist[i] = 0u;
    load_tile_async(tile, heat + (size_t)c * HW, r0, tid);
    __syncthreads();

    const int row = tid >> 3;            // 0..31
    const int seg = (tid & 7) * SEGW;    // 0..336
    const float* rm1 = &tile[row * WW];
    const float* rc  = &tile[(row + 1) * WW];
    const float* rp1 = &tile[(row + 2) * WW];

    float vmL = (seg > 0) ? fmaxf(fmaxf(rm1[seg - 1], rc[seg - 1]), rp1[seg - 1])
                          : -__builtin_inff();
    float vmC = fmaxf(fmaxf(rm1[seg], rc[seg]), rp1[seg]);
#pragma unroll 4
    for (int cix = seg; cix < seg + SEGW; ++cix) {
        float vmR = (cix + 1 < WW)
                        ? fmaxf(fmaxf(rm1[cix + 1], rc[cix + 1]), rp1[cix + 1])
                        : -__builtin_inff();
        float m9 = fmaxf(fmaxf(vmL, vmC), vmR);
        float x = rc[cix];
        if (x >= m9) atomicAdd(&shist[fkey(x) >> BIN_SHIFT], 1u);
        vmL = vmC; vmC = vmR;
    }
    __syncthreads();

    for (int i = tid; i < NBINS; i += 256) {
        unsigned h = shist[i];
        if (h) atomicAdd(&hist[z * NBINS + i], h);
    }
}

// ---------------------------------------------------------------------------
// Pass 2: scan histogram from the top to find the key threshold covering >=K.
// ---------------------------------------------------------------------------
__global__ void find_thresh(const unsigned* __restrict__ hist,
                            unsigned* __restrict__ thresh) {
    int z = blockIdx.x;
    if (threadIdx.x == 0) {
        unsigned cum = 0, tk = 0;
        for (int b = NBINS - 1; b >= 0; --b) {
            cum += hist[z * NBINS + b];
            if (cum >= KTOP) { tk = ((unsigned)b) << BIN_SHIFT; break; }
        }
        thresh[z] = tk;
    }
}

// ---------------------------------------------------------------------------
// Pass 3: re-scan (second pass hits the 192MB L2; both heatmaps = 94MB
// resident) and compact peaks at/above threshold.
// ---------------------------------------------------------------------------
__global__ __launch_bounds__(256) void nms_compact(const float* __restrict__ tlh,
                                                   const float* __restrict__ brh,
                                                   const unsigned* __restrict__ thresh,
                                                   Cand* __restrict__ cand,
                                                   unsigned* __restrict__ cnt) {
    __shared__ __align__(16) float tile[(ROWS + 2) * WW];
    const int tid = threadIdx.x;
    const int z = blockIdx.z;
    const float* heat = z ? brh : tlh;
    const int c = blockIdx.y;
    const int r0 = blockIdx.x * ROWS;
    const unsigned tk = thresh[z];

    load_tile_async(tile, heat + (size_t)c * HW, r0, tid);
    __syncthreads();

    const int row = tid >> 3;
    const int seg = (tid & 7) * SEGW;
    const float* rm1 = &tile[row * WW];
    const float* rc  = &tile[(row + 1) * WW];
    const float* rp1 = &tile[(row + 2) * WW];

    float vmL = (seg > 0) ? fmaxf(fmaxf(rm1[seg - 1], rc[seg - 1]), rp1[seg - 1])
                          : -__builtin_inff();
    float vmC = fmaxf(fmaxf(rm1[seg], rc[seg]), rp1[seg]);
#pragma unroll 4
    for (int cix = seg; cix < seg + SEGW; ++cix) {
        float vmR = (cix + 1 < WW)
                        ? fmaxf(fmaxf(rm1[cix + 1], rc[cix + 1]), rp1[cix + 1])
                        : -__builtin_inff();
        float m9 = fmaxf(fmaxf(vmL, vmC), vmR);
        float x = rc[cix];
        unsigned key = fkey(x);
        if (x >= m9 && key >= tk) {
            unsigned pos = atomicAdd(&cnt[z], 1u);
            if (pos < CAND_CAP) {
                cand[z * CAND_CAP + pos].key = key;
                cand[z * CAND_CAP + pos].idx =
                    (unsigned)(c * HW + (r0 + row) * WW + cix);
            }
        }
        vmL = vmC; vmC = vmR;
    }
}

// ---------------------------------------------------------------------------
// Pass 4: exact top-100 per heatmap (score desc, index asc tie-break, as
// lax.top_k) + gather of class/tag/offset-adjusted coords.
// sel layout per heatmap: [score,cls,xf,yf,tag] x 100
// ---------------------------------------------------------------------------
__global__ __launch_bounds__(256) void select_topk(const Cand* __restrict__ cand,
                                                   const unsigned* __restrict__ cnt,
                                                   const float* __restrict__ tle,
                                                   const float* __restrict__ bre,
                                                   const float* __restrict__ tlo,
                                                   const float* __restrict__ bro,
                                                   float* __restrict__ sel) {
    __shared__ Cand sc[CAND_CAP];
    __shared__ unsigned rk[256];
    __shared__ unsigned ri[256];
    __shared__ int rp[256];
    const int tid = threadIdx.x;
    const int z = blockIdx.x;
    const float* embd = z ? bre : tle;
    const float* offs = z ? bro : tlo;

    unsigned n = cnt[z];
    if (n > CAND_CAP) n = CAND_CAP;
    for (int i = tid; i < CAND_CAP; i += 256) {
        if (i < (int)n) sc[i] = cand[z * CAND_CAP + i];
        else { sc[i].key = 0u; sc[i].idx = 0xFFFFFFFFu; }
    }
    __syncthreads();

    for (int k = 0; k < KTOP; ++k) {
        unsigned bk = 0u, bi = 0xFFFFFFFFu;
        int bp = -1;
        for (int i = tid; i < CAND_CAP; i += 256) {
            Cand e = sc[i];
            if (e.key > bk || (e.key == bk && e.idx < bi)) { bk = e.key; bi = e.idx; bp = i; }
        }
        rk[tid] = bk; ri[tid] = bi; rp[tid] = bp;
        __syncthreads();
        for (int s = 128; s > 0; s >>= 1) {
            if (tid < s) {
                if (rk[tid + s] > rk[tid] ||
                    (rk[tid + s] == rk[tid] && ri[tid + s] < ri[tid])) {
                    rk[tid] = rk[tid + s]; ri[tid] = ri[tid + s]; rp[tid] = rp[tid + s];
                }
            }
            __syncthreads();
        }
        if (tid == 0) {
            float scv = 0.f, clv = 0.f, xfv = 0.f, yfv = 0.f, tgv = 0.f;
            if (rp[0] >= 0 && rk[0] > 0u) {
                sc[rp[0]].key = 0u; sc[rp[0]].idx = 0xFFFFFFFFu;
                float lg = kfloat(rk[0]);
                scv = 1.f / (1.f + expf(-lg));
                unsigned fi = ri[0];
                unsigned ch = fi / HW;
                unsigned rem = fi - ch * HW;
                unsigned yy = rem / WW;
                unsigned xx = rem - yy * WW;
                clv = (float)ch;
                tgv = embd[rem];
                xfv = (float)xx + offs[rem];        // channel 0 = x offset
                yfv = (float)yy + offs[HW + rem];   // channel 1 = y offset
            }
            float* s5 = sel + z * 5 * KTOP;
            s5[0 * KTOP + k] = scv;
            s5[1 * KTOP + k] = clv;
            s5[2 * KTOP + k] = xfv;
            s5[3 * KTOP + k] = yfv;
            s5[4 * KTOP + k] = tgv;
        }
        __syncthreads();
    }
}

// ---------------------------------------------------------------------------
// Pass 5: 100x100 pair matrix via V_WMMA_F32_16X16X4_F32 outer ops:
//   sum  = [s_tl | 1] x [1 ; s_br]   -> s_tl_i + s_br_j   (exact f32)
//   diff = [t_tl | 1] x [1 ; -t_br]  -> t_tl_i - t_br_j   (exact f32)
// then branchless validity mask, pack (score, ~index) into u64 keys and
// bitonic-sort 16K entries in 128KB dynamic LDS (320KB WGP). Descending u64
// order == lax.top_k tie semantics (higher score, then lower flat index).
// ---------------------------------------------------------------------------
__global__ __launch_bounds__(512) void pair_topk(const float* __restrict__ sel,
                                                 float* __restrict__ out) {
    extern __shared__ unsigned long long sb[];   // SORT_N u64 keys
    __shared__ float s_tl[112], c_tl[112], x_tl[112], y_tl[112], t_tl[112];
    __shared__ float s_br[112], c_br[112], x_br[112], y_br[112], t_br[112];
    const int tid = threadIdx.x;

    for (int i = tid; i < 112; i += 512) {
        bool v = i < KTOP;
        s_tl[i] = v ? sel[0 * KTOP + i] : 0.f;
        c_tl[i] = v ? sel[1 * KTOP + i] : -1.f;
        x_tl[i] = v ? sel[2 * KTOP + i] : 0.f;
        y_tl[i] = v ? sel[3 * KTOP + i] : 0.f;
        t_tl[i] = v ? sel[4 * KTOP + i] : 0.f;
        const float* sB = sel + 5 * KTOP;
        s_br[i] = v ? sB[0 * KTOP + i] : 0.f;
        c_br[i] = v ? sB[1 * KTOP + i] : -2.f;
        x_br[i] = v ? sB[2 * KTOP + i] : 0.f;
        y_br[i] = v ? sB[3 * KTOP + i] : 0.f;
        t_br[i] = v ? sB[4 * KTOP + i] : 0.f;
    }
    for (int i = tid; i < SORT_N; i += 512) sb[i] = 0ull;
    __syncthreads();

    const int lane = tid & 31;
    const int wave = tid >> 5;                   // 16 waves
    const bool lo = lane < 16;
    const int l16 = lane & 15;

    for (int t = wave; t < 49; t += 16) {        // 7x7 tiles of 16x16 over 112x112
        int ti = t / 7, tj = t - ti * 7;
        // A is 16x4 (M x K): lanes 0-15 hold K=0,1; lanes 16-31 hold K=2,3
        v2f a, b, at, bt;
        a[0]  = lo ? s_tl[ti * 16 + l16] : 0.f;  // K0 column = tl value
        a[1]  = lo ? 1.f : 0.f;                  // K1 column = ones
        b[0]  = lo ? 1.f : 0.f;                  // B row K0 = ones
        b[1]  = lo ? s_br[tj * 16 + l16] : 0.f;  // B row K1 = br value
        at[0] = lo ? t_tl[ti * 16 + l16] : 0.f;
        at[1] = a[1];
        bt[0] = b[0];
        bt[1] = lo ? -t_br[tj * 16 + l16] : 0.f;
        v8f cz = {};
        v8f sum = __builtin_amdgcn_wmma_f32_16x16x4_f32(false, a,  false, b,
                                                        (short)0, cz, false, false);
        v8f dif = __builtin_amdgcn_wmma_f32_16x16x4_f32(false, at, false, bt,
                                                        (short)0, cz, false, false);
        int jg = tj * 16 + l16;
#pragma unroll
        for (int v = 0; v < 8; ++v) {
            int ig = ti * 16 + v + (lo ? 0 : 8); // C/D layout: lanes>=16 hold M=v+8
            if (ig < KTOP && jg < KTOP) {
                float scv = 0.5f * sum[v];
                float dd = fabsf(dif[v]);
                int inv = (dd > 0.5f) | (c_tl[ig] != c_br[jg]) |
                          (x_tl[ig] > x_br[jg]) | (y_tl[ig] > y_br[jg]);
                if (inv) scv = -1.f;
                unsigned idx = (unsigned)(ig * KTOP + jg);
                sb[idx] = ((unsigned long long)fkey(scv) << 32) |
                          (unsigned long long)(~idx);
            }
        }
    }
    __syncthreads();

    // bitonic sort, descending, SORT_N = 2^14 (padding keys are 0 -> sink)
    for (int kk = 2; kk <= SORT_N; kk <<= 1) {
        for (int jj = kk >> 1; jj > 0; jj >>= 1) {
            for (int i = tid; i < SORT_N; i += 512) {
                int ixj = i ^ jj;
                if (ixj > i) {
                    unsigned long long A = sb[i], B = sb[ixj];
                    bool swp = ((i & kk) == 0) ? (A < B) : (A > B);
                    if (swp) { sb[i] = B; sb[ixj] = A; }
                }
            }
            __syncthreads();
        }
    }

    // emit [5, 1000]: scores, tl_x, tl_y, br_x, br_y
    for (int d = tid; d < NDETS; d += 512) {
        unsigned long long e = sb[d];
        unsigned sk = (unsigned)(e >> 32);
        unsigned idx = ~((unsigned)(e & 0xFFFFFFFFull));
        int ig = idx / KTOP, jg = idx - ig * KTOP;
        out[0 * NDETS + d] = kfloat(sk);
        out[1 * NDETS + d] = x_tl[ig];
        out[2 * NDETS + d] = y_tl[ig];
        out[3 * NDETS + d] = x_br[jg];
        out[4 * NDETS + d] = y_br[jg];
    }
}

extern "C" void kernel_launch(void* const* d_in, const int* in_sizes, int n_in,
                              void* d_out, int out_size, void* d_ws, size_t ws_size,
                              hipStream_t stream) {
    (void)in_sizes; (void)n_in; (void)out_size; (void)ws_size;
    const float* tlh = (const float*)d_in[0];
    const float* brh = (const float*)d_in[1];
    const float* tle = (const float*)d_in[2];
    const float* bre = (const float*)d_in[3];
    const float* tlo = (const float*)d_in[4];
    const float* bro = (const float*)d_in[5];
    float* out = (float*)d_out;

    unsigned char* ws = (unsigned char*)d_ws;
    unsigned* hist   = (unsigned*)(ws);            // 2*2048*4 = 16384 B
    unsigned* cnt    = (unsigned*)(ws + 16384);    // 8 B
    unsigned* thresh = (unsigned*)(ws + 16392);    // 8 B
    float*    sel    = (float*)(ws + 16400);       // 2*5*100*4 = 4000 B
    Cand*     cand   = (Cand*)(ws + 20416);        // 2*6144*8 = 98304 B (8B aligned)

    hipMemsetAsync(ws, 0, 16392, stream);          // zero hist + counters

    dim3 grid(HH / ROWS, CC, 2);
    nms_hist<<<grid, 256, 0, stream>>>(tlh, brh, hist);
    find_thresh<<<2, 32, 0, stream>>>(hist, thresh);
    nms_compact<<<grid, 256, 0, stream>>>(tlh, brh, thresh, cand, cnt);
    select_topk<<<2, 256, 0, stream>>>(cand, cnt, tle, bre, tlo, bro, sel);
    pair_topk<<<1, 512, SORT_N * sizeof(unsigned long long), stream>>>(sel, out);
}